// Bernoulli_22428319220264
// MI455X (gfx1250) — compile-verified
//
#include <hip/hip_runtime.h>
#include <stdint.h>

// Elementwise soft-Bernoulli edge sampling (sigmoid -> logit + logit(u) ->
// sigmoid(./T) -> epsilon threshold). Pure streaming map: 768 MiB of HBM
// traffic => ~33us floor at 23.3 TB/s. No matmul => no WMMA. Input staged via
// the gfx1250 async global->LDS pipeline (ASYNCcnt); outputs via 128-bit
// non-temporal stores. Math uses raw v_rcp/v_exp/v_log TRANS ops to keep the
// VALU cost (~18 ops/element) far below the memory roofline.

typedef float v4f __attribute__((ext_vector_type(4)));
// Pointee type required by __builtin_amdgcn_global_load_async_to_lds_b128
// (per hipcc diagnostic): int __attribute__((vector_size(16)))
typedef int v4i_vs __attribute__((vector_size(4 * sizeof(int))));

#define __AS1 __attribute__((address_space(1)))
#define __AS3 __attribute__((address_space(3)))

#ifndef __has_builtin
#define __has_builtin(x) 0
#endif

#if defined(__HIP_DEVICE_COMPILE__) && defined(__gfx1250__) && \
    __has_builtin(__builtin_amdgcn_global_load_async_to_lds_b128)
#define USE_ASYNC_LDS 1
#else
#define USE_ASYNC_LDS 0
#endif

#if defined(__HIP_DEVICE_COMPILE__) && __has_builtin(__builtin_amdgcn_s_wait_asynccnt)
#define WAIT_ASYNCCNT(n) __builtin_amdgcn_s_wait_asynccnt(n)
#else
#define WAIT_ASYNCCNT(n) asm volatile("s_wait_asynccnt " #n ::: "memory")
#endif

constexpr int BLOCK = 256;  // 8 waves (wave32) per workgroup
constexpr int ITERS = 32;   // float4 chunks per thread

__device__ __forceinline__ void soft_edge4(const v4f x, const float logit_u,
                                           v4f& wv, v4f& mv) {
  constexpr float LOG2E = 1.44269504088896340f;   // 1/ln(2)
  constexpr float LN2 = 0.69314718055994531f;
  constexpr float N2LOG2E = -2.0f * LOG2E;        // fold /T (T=0.5) and -1/ln2
#pragma unroll
  for (int k = 0; k < 4; ++k) {
    const float xi = x[k];
    // s = sigmoid(x) = 1/(1 + exp(-x)); exp(-x) = exp2(-x*log2e)
    const float e = __builtin_amdgcn_exp2f(xi * -LOG2E);
    const float s = __builtin_amdgcn_rcpf(1.0f + e);
    // divide_no_nan(s, 1-s); log(0) -> -inf matches jnp.log(0)
    const float d = 1.0f - s;
    const float r = (d == 0.0f) ? 0.0f : s * __builtin_amdgcn_rcpf(d);
    // l = log(r) + logit(u) = log2(r)*ln2 + logit(u)
    const float l = __builtin_amdgcn_logf(r) * LN2 + logit_u;
    // y = sigmoid(l / T) = 1/(1 + exp2(-2*l*log2e))
    const float y =
        __builtin_amdgcn_rcpf(1.0f + __builtin_amdgcn_exp2f(l * N2LOG2E));
    const bool keep = y > 0.5f;  // EPSILON = 0.5
    wv[k] = keep ? y : 0.0f;
    mv[k] = keep ? 1.0f : 0.0f;
  }
}

__global__ __launch_bounds__(BLOCK) void bernoulli_soft_edges(
    const v4f* __restrict__ in, const float* __restrict__ noise,
    v4f* __restrict__ wout, v4f* __restrict__ mout,
    unsigned long long nvec) {
  const unsigned tid = threadIdx.x;

  // logit of the scalar uniform noise, with divide_no_nan semantics.
  // Once per thread -> precise math is fine here.
  const float u = noise[0];
  const float du = 1.0f - u;
  const float ru = (du == 0.0f) ? 0.0f : u / du;
  const float logit_u = __logf(ru);

  const unsigned long long base =
      (unsigned long long)blockIdx.x * (unsigned long long)(ITERS * BLOCK) + tid;

#if USE_ASYNC_LDS
  // Double-buffered async global->LDS staging. Each thread pulls one float4
  // per iteration into its private LDS slot; only ASYNCcnt ordering is
  // needed (no cross-thread LDS sharing => no barrier).
  __shared__ v4f stage[2][BLOCK];

  const unsigned long long last = nvec ? (nvec - 1) : 0;
  {
    const unsigned long long safe = base < nvec ? base : last;
    __builtin_amdgcn_global_load_async_to_lds_b128(
        (__AS1 v4i_vs*)(in + safe), (__AS3 v4i_vs*)&stage[0][tid], 0, 0);
  }
#pragma unroll 1
  for (int i = 0; i < ITERS; ++i) {
    const unsigned long long idx = base + (unsigned long long)i * BLOCK;
    if (i + 1 < ITERS) {
      const unsigned long long nidx = idx + BLOCK;
      const unsigned long long safe = nidx < nvec ? nidx : last;
      __builtin_amdgcn_global_load_async_to_lds_b128(
          (__AS1 v4i_vs*)(in + safe), (__AS3 v4i_vs*)&stage[(i + 1) & 1][tid], 0, 0);
      WAIT_ASYNCCNT(1);  // chunk i complete, next chunk in flight
    } else {
      WAIT_ASYNCCNT(0);
    }
    asm volatile("" ::: "memory");  // keep LDS read after the wait
    if (idx < nvec) {
      const v4f x = stage[i & 1][tid];
      v4f wv, mv;
      soft_edge4(x, logit_u, wv, mv);
      __builtin_nontemporal_store(wv, wout + idx);
      __builtin_nontemporal_store(mv, mout + idx);
    }
  }
#else
  // Fallback: direct 128-bit NT loads + explicit prefetch of the next chunk.
#pragma unroll 1
  for (int i = 0; i < ITERS; ++i) {
    const unsigned long long idx = base + (unsigned long long)i * BLOCK;
    if (idx < nvec) {
      if (i + 1 < ITERS) {
        const unsigned long long nidx = idx + BLOCK;
        if (nidx < nvec) __builtin_prefetch(in + nidx, 0, 0);
      }
      const v4f x = __builtin_nontemporal_load(in + idx);
      v4f wv, mv;
      soft_edge4(x, logit_u, wv, mv);
      __builtin_nontemporal_store(wv, wout + idx);
      __builtin_nontemporal_store(mv, mout + idx);
    }
  }
#endif
}

extern "C" void kernel_launch(void* const* d_in, const int* in_sizes, int n_in,
                              void* d_out, int out_size, void* d_ws, size_t ws_size,
                              hipStream_t stream) {
  (void)n_in; (void)out_size; (void)d_ws; (void)ws_size;

  const float* sim = (const float*)d_in[0];     // N*N float32
  const float* noise = (const float*)d_in[1];   // 1 float32

  const unsigned long long n = (unsigned long long)in_sizes[0];  // N*N
  float* w = (float*)d_out;       // weights: first N*N floats
  float* m = w + n;               // mask (0/1): second N*N floats

  const unsigned long long nvec = n / 4;  // N*N is divisible by 4
  const unsigned long long per_block = (unsigned long long)BLOCK * (unsigned long long)ITERS;
  const unsigned grid = (unsigned)((nvec + per_block - 1) / per_block);

  bernoulli_soft_edges<<<grid, BLOCK, 0, stream>>>(
      (const v4f*)sim, noise, (v4f*)w, (v4f*)m, nvec);
}